// ScalableVIN_31224412242002
// MI455X (gfx1250) — compile-verified
//
#include <hip/hip_runtime.h>

// ---- problem constants (match reference) ----
#define B 8
#define P 2
#define A 8
#define T 8
#define H 32
#define W 32
#define K3 27
#define KK 54            // P*K3: reduction length == softmax length
#define KPAD 56          // K padded to multiple of 4 -> 14 wmma k-steps
#define NKS (KPAD/4)     // 14 wmma k-steps
#define NGROUP 16        // A*P softmax groups
#define MROWS 64         // 54 rows padded to 4 M-tiles of 16
#define DEPTH 4
#define CELLS_PER_B (T*H*W)          // 8192
#define NCELL (B*CELLS_PER_B)        // 65536
#define NCOL 128                      // columns per workgroup (8 N-tiles)

// padded rv dims (pad=1)
#define PT (T+2)
#define PH (H+2)
#define PW (W+2)
#define RV_I_STRIDE (PT*PH*PW)        // 11560
#define RV_B_STRIDE (P*RV_I_STRIDE)   // 23120
#define RV_TOTAL (B*RV_B_STRIDE)      // 184960

// LDS strides: even (8B-alignment for v2f loads) and conflict-aware (64 banks)
#define WSTR 58
#define USTR 58
#define LSTR 130

#define W_ELEMS (MROWS * WSTR)        // 3712
#define U_ELEMS (NCOL * USTR)         // 7424
#define L_ELEMS (MROWS * LSTR)        // 8320
#define Q_ELEMS (NGROUP * NCOL)       // 2048
#define SMEM_FLOATS (W_ELEMS + U_ELEMS + L_ELEMS + Q_ELEMS)   // 21504 -> 86016 B

typedef __attribute__((ext_vector_type(2))) float v2f;
typedef __attribute__((ext_vector_type(8))) float v8f;

// rvp[b][i][tt][hh][ww] = (interior ? values+rewards : 0)
__global__ void vin_pad_rv(const float* __restrict__ v, const float* __restrict__ r,
                           float* __restrict__ rvp) {
  int idx = blockIdx.x * blockDim.x + threadIdx.x;
  if (idx >= RV_TOTAL) return;
  int ww = idx % PW;
  int hh = (idx / PW) % PH;
  int tt = (idx / (PW * PH)) % PT;
  int bi = idx / RV_I_STRIDE;          // b*P + i
  float val = 0.0f;
  if (tt >= 1 && tt <= T && hh >= 1 && hh <= H && ww >= 1 && ww <= W) {
    int src = bi * CELLS_PER_B + (tt - 1) * (H * W) + (hh - 1) * W + (ww - 1);
    val = v[src] + r[src];
  }
  rvp[idx] = val;
}

// One VI step: implicit GEMM (W2[864x54] x U[54xN]) + fused softmax·dot + action max.
// w2 = trans_w viewed flat as [CO=864][KK=54] (already in that memory order).
__launch_bounds__(256)
__global__ void vin_step(const float* __restrict__ rvp,
                         const float* __restrict__ w2,
                         float* __restrict__ out) {   // [B][P][T][H][W] slice
  extern __shared__ __attribute__((aligned(16))) float smem[];
  float* w_lds = smem;                  // [MROWS][WSTR] one group's weights (padded)
  float* u_lds = w_lds + W_ELEMS;       // [NCOL][USTR]  unfold values, column-major
  float* L_lds = u_lds + U_ELEMS;       // [MROWS][LSTR] logits tile
  float* q_lds = L_lds + L_ELEMS;       // [NGROUP][NCOL] softmax·dot results

  const int tid = threadIdx.x;
  const int n0 = blockIdx.x * NCOL;

  // ---- gather u: u_lds[c][j] = rvp[b][i][t+di][h+dj][w+dk], j = i*27 + di*9+dj*3+dk
  for (int idx = tid; idx < U_ELEMS; idx += 256) {
    int c = idx / USTR;
    int j = idx % USTR;
    float val = 0.0f;
    if (j < KK) {
      int n = n0 + c;
      int b  = n >> 13;
      int rm = n & 8191;
      int t  = rm >> 10;
      int h  = (rm >> 5) & 31;
      int w  = rm & 31;
      int i  = j / K3;
      int k  = j - i * K3;
      int di = k / 9, dj = (k / 3) % 3, dk = k % 3;
      val = rvp[b * RV_B_STRIDE + i * RV_I_STRIDE
                + (t + di) * (PH * PW) + (h + dj) * PW + (w + dk)];
    }
    u_lds[idx] = val;
  }

  const int lane = tid & 31;   // wave32
  const int wave = tid >> 5;   // 8 waves
  const int half = lane >> 4;
  const int l16  = lane & 15;
  const int bcol = wave * 16 + l16;   // this wave's N-tile column (nt == wave)

  for (int o = 0; o < NGROUP; ++o) {
    __syncthreads();  // prior group's softmax done; u gather done on first pass
    // ---- stage this group's 54x54 weight block (zero-padded to 64x56)
    for (int idx = tid; idx < W_ELEMS; idx += 256) {
      int row = idx / WSTR;
      int k   = idx % WSTR;
      float val = 0.0f;
      if (row < KK && k < KK) val = w2[(o * KK + row) * KK + k];
      w_lds[idx] = val;
    }
    if (o + 1 < NGROUP && tid < 64)
      __builtin_prefetch(&w2[(o + 1) * KK * KK + tid * 48], 0, 0);  // global_prefetch_b8
    __syncthreads();

    // ---- preload all 14 B-operand pairs for this wave's N-tile (group-invariant)
    v2f breg[NKS];
#pragma unroll
    for (int ks = 0; ks < NKS; ++ks)
      breg[ks] = *(const v2f*)&u_lds[bcol * USTR + ks * 4 + half * 2];

    // ---- GEMM: M-tiles processed in pairs with two independent accumulators so
    //      consecutive v_wmma ops have no RAW chain and A-loads stay pipelined.
#pragma unroll
    for (int mp = 0; mp < 2; ++mp) {
      const int r0 = (mp * 2 + 0) * 16 + l16;
      const int r1 = (mp * 2 + 1) * 16 + l16;
      v8f acc0 = {}, acc1 = {};
#pragma unroll
      for (int ks = 0; ks < NKS; ++ks) {
        int k0 = ks * 4 + half * 2;
        v2f a0 = *(const v2f*)&w_lds[r0 * WSTR + k0];  // 8B-aligned: WSTR,k0 even
        v2f a1 = *(const v2f*)&w_lds[r1 * WSTR + k0];
        acc0 = __builtin_amdgcn_wmma_f32_16x16x4_f32(
            false, a0, false, breg[ks], (short)0, acc0, false, false);
        acc1 = __builtin_amdgcn_wmma_f32_16x16x4_f32(
            false, a1, false, breg[ks], (short)0, acc1, false, false);
      }
      // D 16x16 f32 layout: VGPR d -> row d (lanes 0-15) / row d+8 (lanes 16-31)
#pragma unroll
      for (int d = 0; d < 8; ++d) {
        int row0 = (mp * 2 + 0) * 16 + d + half * 8;
        int row1 = (mp * 2 + 1) * 16 + d + half * 8;
        L_lds[row0 * LSTR + wave * 16 + l16] = acc0[d];
        L_lds[row1 * LSTR + wave * 16 + l16] = acc1[d];
      }
    }
    __syncthreads();

    // ---- fused softmax over 54 logits + dot with u: 2 threads per column,
    //      shfl_xor(1) pair-reduction (pair lanes are adjacent within a wave).
    {
      const int c = tid >> 1;        // 0..127
      const int r = tid & 1;
      float m = -3.4e38f;
      for (int j = r; j < KK; j += 2) m = fmaxf(m, L_lds[j * LSTR + c]);
      m = fmaxf(m, __shfl_xor(m, 1, 32));
      float s = 0.0f, q = 0.0f;
      for (int j = r; j < KK; j += 2) {
        float e = __expf(L_lds[j * LSTR + c] - m);
        s += e;
        q += e * u_lds[c * USTR + j];
      }
      s += __shfl_xor(s, 1, 32);
      q += __shfl_xor(q, 1, 32);
      if (r == 0) q_lds[o * NCOL + c] = q / s;
    }
  }
  __syncthreads();

  // ---- action max: o = p*A + a ; write layer slice [b][p][t][h][w]
  {
    int c = tid & (NCOL - 1);
    int p = tid >> 7;
    float m = -3.4e38f;
#pragma unroll
    for (int a = 0; a < A; ++a) m = fmaxf(m, q_lds[(p * A + a) * NCOL + c]);
    int n  = n0 + c;
    int b  = n >> 13;
    int rm = n & 8191;
    out[b * (P * CELLS_PER_B) + p * CELLS_PER_B + rm] = m;
  }
}

extern "C" void kernel_launch(void* const* d_in, const int* in_sizes, int n_in,
                              void* d_out, int out_size, void* d_ws, size_t ws_size,
                              hipStream_t stream) {
  const float* values  = (const float*)d_in[0];   // [B,P,T,H,W]
  const float* rewards = (const float*)d_in[1];   // [B,P,T,H,W]
  const float* trans_w = (const float*)d_in[2];   // [864,2,3,3,3] == [864][54] flat
  float* out = (float*)d_out;                     // [DEPTH,B,P,T,H,W]
  float* rvp = (float*)d_ws;                      // padded rv scratch (740 KB)

  const size_t step_elems = (size_t)B * P * CELLS_PER_B;  // 131072
  const size_t smem_bytes = SMEM_FLOATS * sizeof(float);  // 86016 B dynamic LDS
  for (int s = 0; s < DEPTH; ++s) {
    const float* vcur = (s == 0) ? values : out + (size_t)(s - 1) * step_elems;
    vin_pad_rv<<<(RV_TOTAL + 255) / 256, 256, 0, stream>>>(vcur, rewards, rvp);
    vin_step<<<NCELL / NCOL, 256, smem_bytes, stream>>>(rvp, trans_w,
                                                        out + (size_t)s * step_elems);
  }
}